// MixUp1D_88356067213415
// MI455X (gfx1250) — compile-verified
//
#include <hip/hip_runtime.h>

// MixUp1D for MI455X (gfx1250).
// x: (256,16,10000) fp32 -> 163.84 MB. Pure streaming blend, BW-bound.
// Strategy:
//  - float4 (b128) loads/stores everywhere (C*L = 160000 % 4 == 0).
//  - Direct x stream: regular RT loads so x stays resident in the 192MB L2.
//  - Permuted x stream: CDNA5 async global->LDS b128 copy (ASYNCcnt path),
//    then ds_load_b128 back. Bandwidth-neutral, exercises the gfx1250 async
//    data mover path.
//  - Output: non-temporal stores (th:TH_STORE_NT) -- write-once stream must
//    not evict x from L2, so the permuted second read of x hits L2.
//  - blockIdx.y = batch index b => perm/lam/apply_rand are wave-uniform
//    (scalar loads), no per-lane integer division.

typedef __attribute__((ext_vector_type(4))) float v4f;

#define B_   256
#define C_   16
#define L_   10000
#define P_   0.5f
#define ROWV4 ((C_ * L_) / 4)   // 40000 float4 per batch item

__global__ __launch_bounds__(256) void mixup_x_kernel(
    const float* __restrict__ x,
    const int*   __restrict__ perm,
    const float* __restrict__ apply_rand,
    const float* __restrict__ lam,
    float*       __restrict__ out)
{
    __shared__ v4f tile[256];

    const int b   = blockIdx.y;
    const int idx = blockIdx.x * blockDim.x + threadIdx.x;

    // Wave-uniform per-batch scalars.
    const int   pb  = perm[b];
    const float ar  = apply_rand[b];
    const float le  = (ar < P_) ? lam[b] : 1.0f;
    const float ole = 1.0f - le;

    if (idx >= ROWV4) return;

    const v4f* __restrict__ xb  = (const v4f*)x + (size_t)b  * ROWV4;
    const v4f* __restrict__ xpb = (const v4f*)x + (size_t)pb * ROWV4;
    v4f*       __restrict__ ob  = (v4f*)out     + (size_t)b  * ROWV4;

    // Direct stream: regular (RT) cached load -- keep x hot in L2.
    v4f xv = xb[idx];

    // Permuted stream through the async global->LDS engine (ASYNCcnt).
    // Each lane copies its own 16B: LDS slot = tile[threadIdx.x].
    {
        unsigned   ldsOff = (unsigned)(size_t)(&tile[threadIdx.x]);
        const v4f* gsrc   = xpb + idx;
        asm volatile("global_load_async_to_lds_b128 %0, %1, off"
                     :
                     : "v"(ldsOff), "v"(gsrc)
                     : "memory");
        asm volatile("s_wait_asynccnt 0" ::: "memory");
    }
    v4f xpv = tile[threadIdx.x];

    v4f r;
    r.x = le * xv.x + ole * xpv.x;
    r.y = le * xv.y + ole * xpv.y;
    r.z = le * xv.z + ole * xpv.z;
    r.w = le * xv.w + ole * xpv.w;

    // Non-temporal store: don't pollute L2 with the write-once output.
    __builtin_nontemporal_store(r, &ob[idx]);
}

__global__ __launch_bounds__(256) void mixup_y_kernel(
    const float* __restrict__ y,
    const int*   __restrict__ perm,
    const float* __restrict__ apply_rand,
    const float* __restrict__ lam,
    float*       __restrict__ out)
{
    const int i = blockIdx.x * blockDim.x + threadIdx.x;
    if (i >= B_ * 6) return;

    const int b = i / 6;
    const int c = i - b * 6;

    const int   pb  = perm[b];
    const float le  = (apply_rand[b] < P_) ? lam[b] : 1.0f;
    const float ole = 1.0f - le;

    const float r = le * y[i] + ole * y[pb * 6 + c];
    __builtin_nontemporal_store(r, &out[i]);
}

extern "C" void kernel_launch(void* const* d_in, const int* in_sizes, int n_in,
                              void* d_out, int out_size, void* d_ws, size_t ws_size,
                              hipStream_t stream) {
    const float* x          = (const float*)d_in[0];
    const float* y          = (const float*)d_in[1];
    const int*   perm       = (const int*)  d_in[2];
    const float* apply_rand = (const float*)d_in[3];
    const float* lam        = (const float*)d_in[4];

    float* out_x = (float*)d_out;
    float* out_y = out_x + (size_t)B_ * C_ * L_;   // 40,960,000 floats in

    dim3 gridX((ROWV4 + 255) / 256, B_, 1);
    mixup_x_kernel<<<gridX, dim3(256, 1, 1), 0, stream>>>(x, perm, apply_rand, lam, out_x);

    const int ny = B_ * 6;
    mixup_y_kernel<<<(ny + 255) / 256, 256, 0, stream>>>(y, perm, apply_rand, lam, out_y);
}